// Conv_13589276525053
// MI455X (gfx1250) — compile-verified
//
#include <hip/hip_runtime.h>

// GCN layer: agg = x + scatter_add(x[sources] -> targets); h = norm*agg; out = h @ W
// N=200000 nodes, C=256 channels, E=600000 edges. fp32 throughout (memory-bound
// GEMM -> fp32 WMMA 16x16x4 keeps exact precision at the same roofline).

#define N_NODES 200000
#define N_EDGES 600000
#define CCH 256

typedef __attribute__((ext_vector_type(2))) float v2f;
typedef __attribute__((ext_vector_type(8))) float v8f;

// ---------------- Phase 1a: agg = x (float4 copy) ----------------
__global__ __launch_bounds__(256) void init_agg(const float4* __restrict__ x4,
                                                float4* __restrict__ agg4) {
    int gid = blockIdx.x * 256 + threadIdx.x;       // N*C/4 = 12.8M threads, exact
    agg4[gid] = x4[gid];
}

// ---------------- Phase 1b: WT[n][k] = W[k][n] ----------------
__global__ __launch_bounds__(256) void transpose_w(const float* __restrict__ W,
                                                   float* __restrict__ WT) {
    int k = blockIdx.x;        // 0..255
    int n = threadIdx.x;       // 0..255
    WT[n * CCH + k] = W[k * CCH + n];
}

// ---------------- Phase 2: scatter-add (64 lanes per edge, float4 gather) ------
__global__ __launch_bounds__(256) void scatter_add(const float4* __restrict__ x4,
                                                   const int* __restrict__ sources,
                                                   const int* __restrict__ targets,
                                                   float* __restrict__ agg) {
    int gid  = blockIdx.x * 256 + threadIdx.x;      // E*64 threads, exact
    int e    = gid >> 6;
    int lane = gid & 63;
    int src = sources[e];
    int tgt = targets[e];
    float4 v = x4[(size_t)src * 64 + lane];         // coalesced 1KB row gather
    float* dst = agg + (size_t)tgt * CCH + lane * 4;
    atomicAdd(dst + 0, v.x);                        // global_atomic_add_f32 (L2-resident)
    atomicAdd(dst + 1, v.y);
    atomicAdd(dst + 2, v.z);
    atomicAdd(dst + 3, v.w);
}

// ---------------- Phase 3: out = (norm*agg) @ W via V_WMMA_F32_16X16X4_F32 -----
// Block: 256 threads = 8 waves; one 16-row tile per block; wave w covers
// output columns [32w, 32w+32) as two 16x16 D tiles. A tile (norm-scaled) is
// staged in LDS with stride 260 floats (260 % 64 == 4) so the per-step b64
// A-fragment reads are LDS bank-conflict-free across all 32 lanes.
#define A_STRIDE 260

__global__ __launch_bounds__(256) void gemm_wmma(const float* __restrict__ agg,
                                                 const float* __restrict__ norm,
                                                 const float* __restrict__ WT,
                                                 float* __restrict__ out) {
    __shared__ float As[16 * A_STRIDE];

    const int tid = threadIdx.x;
    const int r0  = blockIdx.x * 16;                 // row-tile base (12500 tiles)

    // Stage norm-scaled A tile: 16 rows x 256 cols, float4 per thread x 4 iters.
    {
        const int row0 = tid >> 6;                   // 0..3
        const int col  = (tid & 63) * 4;             // 0..252
        #pragma unroll
        for (int i = 0; i < 4; ++i) {
            const int rr = row0 + i * 4;             // 0..15
            const float s = norm[r0 + rr];
            const float4 v = *(const float4*)(agg + (size_t)(r0 + rr) * CCH + col);
            float4 w;
            w.x = v.x * s; w.y = v.y * s; w.z = v.z * s; w.w = v.w * s;
            *(float4*)(&As[rr * A_STRIDE + col]) = w; // rr*1040B is 16B-aligned
        }
    }
    __syncthreads();

    const int wave = tid >> 5;                       // 0..7
    const int lane = tid & 31;
    const int m    = lane & 15;                      // A row / D col within tile
    const int kh   = (lane >> 4) * 2;                // K sub-pair select (0 or 2)
    const int n0   = wave * 32;                      // first column tile
    const int n1   = n0 + 16;                        // second column tile

    const float* __restrict__ b0p = WT + (size_t)(n0 + m) * CCH + kh;
    const float* __restrict__ b1p = WT + (size_t)(n1 + m) * CCH + kh;
    const float* __restrict__ ap  = &As[m * A_STRIDE + kh];

    v8f acc0 = {};
    v8f acc1 = {};

    #pragma unroll 8
    for (int k = 0; k < CCH; k += 4) {
        v2f a  = *(const v2f*)(ap + k);              // ds_load_b64, conflict-free
        v2f b0 = *(const v2f*)(b0p + k);             // global_load_b64 (WT hot in L2)
        v2f b1 = *(const v2f*)(b1p + k);
        acc0 = __builtin_amdgcn_wmma_f32_16x16x4_f32(false, a, false, b0,
                                                     (short)0, acc0, false, false);
        acc1 = __builtin_amdgcn_wmma_f32_16x16x4_f32(false, a, false, b1,
                                                     (short)0, acc1, false, false);
    }

    // D layout: VGPR j -> row (j + 8*(lane>>4)), col (lane&15) within tile.
    float* __restrict__ o = out + (size_t)r0 * CCH;
    const int rbase = (lane >> 4) * 8;
    const int c0 = n0 + (lane & 15);
    #pragma unroll
    for (int j = 0; j < 8; ++j) {
        const size_t roff = (size_t)(rbase + j) * CCH;
        o[roff + c0]      = acc0[j];
        o[roff + c0 + 16] = acc1[j];
    }
}

extern "C" void kernel_launch(void* const* d_in, const int* in_sizes, int n_in,
                              void* d_out, int out_size, void* d_ws, size_t ws_size,
                              hipStream_t stream) {
    const float* x       = (const float*)d_in[0];
    const int*   sources = (const int*)d_in[1];
    const int*   targets = (const int*)d_in[2];
    const float* norm    = (const float*)d_in[3];
    const float* W       = (const float*)d_in[4];
    float*       out     = (float*)d_out;

    float* agg = (float*)d_ws;                       // N*C floats = 204.8 MB
    float* WT  = agg + (size_t)N_NODES * CCH;        // + 256 KB

    init_agg<<<(N_NODES * CCH / 4) / 256, 256, 0, stream>>>(
        (const float4*)x, (float4*)agg);
    transpose_w<<<CCH, CCH, 0, stream>>>(W, WT);
    scatter_add<<<(N_EDGES * 64) / 256, 256, 0, stream>>>(
        (const float4*)x, sources, targets, agg);
    gemm_wmma<<<N_NODES / 16, 256, 0, stream>>>(agg, norm, WT, out);
}